// BaseBuilder_44152263803427
// MI455X (gfx1250) — compile-verified
//
#include <hip/hip_runtime.h>

typedef __attribute__((ext_vector_type(2))) float v2f;
typedef __attribute__((ext_vector_type(4))) float v4f;
typedef __attribute__((ext_vector_type(8))) float v8f;
typedef __attribute__((ext_vector_type(4))) int   v4i;

typedef __attribute__((address_space(1))) v4i g_v4i;   // global int4
typedef __attribute__((address_space(3))) v4i l_v4i;   // LDS int4

#define TILE_K     32
#define LDS_STRIDE 36      // 32 + 4 pad: rows stay 16B-aligned; frag reads are bank-conflict-free
#define BLOCK      32      // one wave32 per workgroup (barriers become S_NOP)
#define M_TILE     16
#define N_TOTAL    64
#define K_TOTAL    1024
#define M_TOTAL    2048
#define ASYNC_OPS  20      // 4 (Q tile) + 16 (P tile) b128 async copies per lane per chunk

#if __has_builtin(__builtin_amdgcn_global_load_async_to_lds_b128)
#define HAVE_ASYNC 1
#else
#define HAVE_ASYNC 0
#endif

#if __has_builtin(__builtin_amdgcn_s_wait_asynccnt)
#define WAIT_ASYNC(n) __builtin_amdgcn_s_wait_asynccnt(n)
#else
#define WAIT_ASYNC(n) asm volatile("s_wait_asynccnt %0" ::"i"(n) : "memory")
#endif

// Copy 16 bytes global -> LDS, async (tracked by ASYNCcnt).
__device__ __forceinline__ void cp_async16(const float* g, float* l) {
#if HAVE_ASYNC
    __builtin_amdgcn_global_load_async_to_lds_b128((g_v4i*)g, (l_v4i*)l, 0, 0);
#else
    *(v4f*)l = *(const v4f*)g;
#endif
}

// Stage one K-chunk: Q tile 16x32 (128 b128 units), P tile 64x32 (512 b128 units).
__device__ __forceinline__ void prefetch_chunk(const float* __restrict__ query,
                                               const float* __restrict__ proto,
                                               float* qs, float* ps,
                                               int lane, int mBase, int k0) {
#pragma unroll
    for (int i = 0; i < 4; ++i) {
        int u = lane + i * 32, row = u >> 3, c4 = (u & 7) << 2;
        cp_async16(query + (size_t)(mBase + row) * K_TOTAL + k0 + c4,
                   qs + row * LDS_STRIDE + c4);
    }
#pragma unroll
    for (int i = 0; i < 16; ++i) {
        int u = lane + i * 32, row = u >> 3, c4 = (u & 7) << 2;
        cp_async16(proto + (size_t)row * K_TOTAL + k0 + c4,
                   ps + row * LDS_STRIDE + c4);
    }
}

// out[i,j] = ||Q[i,:] - P[j,:]||^2 = qnorm[i] + pnorm[j] - 2*(Q @ P^T)[i,j]
__global__ __launch_bounds__(BLOCK) void l2dist_wmma_f32(
    const float* __restrict__ proto,   // [64, 1024]
    const float* __restrict__ query,   // [2048, 1024]
    float* __restrict__ out)           // [2048, 64]
{
    __shared__ float Qs[2][M_TILE * LDS_STRIDE];   // 2 x 2.25 KB
    __shared__ float Ps[2][N_TOTAL * LDS_STRIDE];  // 2 x 9 KB

    const int lane  = threadIdx.x & 31;
    const int half  = lane >> 4;        // 0: K%4 in {0,1}, 1: K%4 in {2,3}
    const int l15   = lane & 15;
    const int mBase = blockIdx.x * M_TILE;

    v8f   c[4] = {};                    // 4 N-tiles of 16x16 f32 accumulators
    float anorm = 0.0f;
    float bnorm[4] = {0.f, 0.f, 0.f, 0.f};

    // Pipeline prologue: chunk 0 into buffer 0.
    prefetch_chunk(query, proto, Qs[0], Ps[0], lane, mBase, 0);

    int buf = 0;
    for (int k0 = 0; k0 < K_TOTAL; k0 += TILE_K) {
        const int nxt = buf ^ 1;
        if (k0 + TILE_K < K_TOTAL) {
            // Issue next chunk, then wait until only those ASYNC_OPS remain:
            // guarantees the current buffer's copies have landed in LDS.
            prefetch_chunk(query, proto, Qs[nxt], Ps[nxt], lane, mBase, k0 + TILE_K);
            WAIT_ASYNC(ASYNC_OPS);
        } else {
            WAIT_ASYNC(0);
        }

        const float* qs = Qs[buf];
        const float* ps = Ps[buf];
#pragma unroll
        for (int kk = 0; kk < TILE_K / 4; ++kk) {
            const int kc = kk * 4 + half * 2;
            // A fragment (16x4 f32): lane<16 -> K%4 in {0,1}, lane>=16 -> {2,3}.
            v2f a = *(const v2f*)(qs + l15 * LDS_STRIDE + kc);
            anorm += a.x * a.x + a.y * a.y;
#pragma unroll
            for (int t = 0; t < 4; ++t) {
                // B fragment (4x16 f32): mirrored per-lane K pair over P rows.
                v2f b = *(const v2f*)(ps + (t * 16 + l15) * LDS_STRIDE + kc);
                bnorm[t] += b.x * b.x + b.y * b.y;
                // (neg_a, A, neg_b, B, c_mod, C, reuse_a, reuse_b)
                c[t] = __builtin_amdgcn_wmma_f32_16x16x4_f32(
                        false, a, false, b, (short)0, c[t], false, false);
            }
        }
        buf = nxt;
    }

    // Combine half-wave K slices: lane then holds the full norm for row (lane & 15).
    float qn = anorm + __shfl_xor(anorm, 16, 32);
    float pn[4];
#pragma unroll
    for (int t = 0; t < 4; ++t)
        pn[t] = bnorm[t] + __shfl_xor(bnorm[t], 16, 32);

    // C layout: VGPR r -> (M = r + 8*half, N = lane&15).
#pragma unroll
    for (int r = 0; r < 8; ++r) {
        int   rowl = r + half * 8;
        float qr   = __shfl(qn, rowl, 32);   // broadcast qnorm for this output row
        int   gi   = (mBase + rowl) * N_TOTAL + l15;
#pragma unroll
        for (int t = 0; t < 4; ++t) {
            out[gi + t * 16] = qr + pn[t] - 2.0f * c[t][r];
        }
    }
}

extern "C" void kernel_launch(void* const* d_in, const int* in_sizes, int n_in,
                              void* d_out, int out_size, void* d_ws, size_t ws_size,
                              hipStream_t stream) {
    (void)in_sizes; (void)n_in; (void)d_ws; (void)ws_size; (void)out_size;
    const float* proto = (const float*)d_in[0];   // [1, 64, 1024] f32
    const float* query = (const float*)d_in[1];   // [1, 2048, 1024] f32
    // d_in[2] (query_idx) and d_in[3] (emb_dim) only determine shapes; unused.
    float* out = (float*)d_out;                   // [2048, 64] f32

    dim3 grid(M_TOTAL / M_TILE);   // 128 single-wave workgroups -> spreads across WGPs/HBM
    dim3 block(BLOCK);             // 32 threads = 1 wave (wave32)
    l2dist_wmma_f32<<<grid, block, 0, stream>>>(proto, query, out);
}